// GATBasedMolecularGraphNeuralNetworkAndEmd_54872502173928
// MI455X (gfx1250) — compile-verified
//
#include <hip/hip_runtime.h>
#include <hip/hip_bf16.h>

// ---------------------------------------------------------------------------
// GATv2 GNN stack for MI455X (gfx1250, wave32).
// Dense transforms use V_WMMA_F32_16X16X4_F32 (fp32 in/out, matches reference
// precision; the workload is edge-gather memory bound, so the fp32 matrix rate
// is not the bottleneck). Edge softmax/aggregation via ordered-uint atomicMax
// + f32 atomicAdd. Everything launched on `stream`; scratch carved from d_ws.
// ---------------------------------------------------------------------------

typedef __attribute__((ext_vector_type(2))) float v2f;
typedef __attribute__((ext_vector_type(8))) float v8f;

#define HIDC 128

static __device__ __forceinline__ unsigned fenc(float f) {
    unsigned u = __float_as_uint(f);
    return (u & 0x80000000u) ? ~u : (u | 0x80000000u);
}
static __device__ __forceinline__ float fdec(unsigned u) {
    u = (u & 0x80000000u) ? (u & 0x7FFFFFFFu) : ~u;
    return __uint_as_float(u);
}
#define NEG_INF_KEY 0x007FFFFFu  // fenc(-inf)

// ---------------------------------------------------------------------------
// GEMM: Out[M x NC] = A[M x K] @ W[K x NC] (+bias, optional leaky-relu 0.01)
// K, NC compile-time so the accumulator tiles stay register-resident and the
// WMMA loops fully unroll. Block = 128 threads (4 waves), tile = 64 x NC.
// Each wave owns 16 rows and NC/16 16x16 fp32 accumulators.
// W staged in LDS K-pair interleaved: element (k,n) -> sW[(k/2)*2NC + 2n + k%2]
// so each B fragment is one aligned b64 LDS load.
// ---------------------------------------------------------------------------
template <int K, int NC>
__launch_bounds__(128)
__global__ void gemm_f32_wmma(const float* __restrict__ A,
                              const float* __restrict__ W,
                              const float* __restrict__ bias,
                              float* __restrict__ Out,
                              int M, int act) {
    constexpr int NT = NC / 16;
    __shared__ float sW[K * NC];   // K-pair interleaved
    __shared__ float sH[64 * K];   // row-major 64-row tile

    const int tid     = threadIdx.x;
    const int rowbase = blockIdx.x * 64;

    for (int i = tid; i < K * NC; i += 128) {
        int k = i / NC, n = i - k * NC;
        sW[(k >> 1) * (2 * NC) + n * 2 + (k & 1)] = W[i];
    }
    for (int i = tid; i < 64 * K; i += 128) {
        int r = i / K, c = i - r * K;
        int gr = rowbase + r;
        sH[i] = (gr < M) ? A[(size_t)gr * K + c] : 0.0f;
    }
    __syncthreads();

    const int wave = tid >> 5;
    const int lane = tid & 31;
    const int half = lane >> 4;   // 0: lanes 0-15, 1: lanes 16-31
    const int lm   = lane & 15;
    const int wrow = wave * 16;

    v8f acc[NT] = {};

    const float* aRow = &sH[(wrow + lm) * K];

#pragma unroll 4
    for (int k0 = 0; k0 < K; k0 += 4) {
        const int ka = k0 + 2 * half;            // even
        v2f av = *(const v2f*)&aRow[ka];         // A[m][ka], A[m][ka+1]
        const float* bRow = &sW[(ka >> 1) * (2 * NC)];
#pragma unroll
        for (int nt = 0; nt < NT; ++nt) {
            v2f bv = *(const v2f*)&bRow[(nt * 16 + lm) * 2];  // B[ka][n],B[ka+1][n]
            acc[nt] = __builtin_amdgcn_wmma_f32_16x16x4_f32(
                false, av, false, bv, (short)0, acc[nt], false, false);
        }
    }

#pragma unroll
    for (int nt = 0; nt < NT; ++nt) {
        int col = nt * 16 + lm;
        float bsum = bias ? bias[col] : 0.0f;
#pragma unroll
        for (int v = 0; v < 8; ++v) {
            int gr = rowbase + wrow + v + 8 * half;  // C: VGPR v -> M=v / v+8
            if (gr < M) {
                float val = acc[nt][v] + bsum;
                if (act) val = val > 0.0f ? val : 0.01f * val;
                Out[(size_t)gr * NC + col] = val;
            }
        }
    }
}

// ---------------------------------------------------------------------------
// Edge kernels. Edge k: (s,d) = (src[k],dst[k]) for k<E, else self loop k-E.
// ---------------------------------------------------------------------------
__launch_bounds__(256)
__global__ void edge_score(const float4* __restrict__ hl4,
                           const float4* __restrict__ hr4,
                           const float4* __restrict__ att4,
                           const int* __restrict__ src,
                           const int* __restrict__ dst,
                           float* __restrict__ e,
                           unsigned* __restrict__ mkey,
                           int E, int EN) {
    int w    = blockIdx.x * (blockDim.x >> 5) + (threadIdx.x >> 5);
    int lane = threadIdx.x & 31;
    if (w >= EN) return;  // uniform per wave
    int s = (w < E) ? src[w] : (w - E);
    int d = (w < E) ? dst[w] : (w - E);
    float4 l = hl4[(size_t)s * 32 + lane];
    float4 r = hr4[(size_t)d * 32 + lane];
    float4 a = att4[lane];
    float x0 = l.x + r.x, x1 = l.y + r.y, x2 = l.z + r.z, x3 = l.w + r.w;
    x0 = x0 > 0.0f ? x0 : 0.2f * x0;
    x1 = x1 > 0.0f ? x1 : 0.2f * x1;
    x2 = x2 > 0.0f ? x2 : 0.2f * x2;
    x3 = x3 > 0.0f ? x3 : 0.2f * x3;
    float p = a.x * x0 + a.y * x1 + a.z * x2 + a.w * x3;
    for (int o = 16; o > 0; o >>= 1) p += __shfl_xor(p, o, 32);
    if (lane == 0) {
        e[w] = p;
        atomicMax(&mkey[d], fenc(p));
    }
}

__global__ void edge_exp(const int* __restrict__ dst,
                         float* __restrict__ e,
                         const unsigned* __restrict__ mkey,
                         float* __restrict__ denom,
                         int E, int EN) {
    int k = blockIdx.x * blockDim.x + threadIdx.x;
    if (k >= EN) return;
    int d = (k < E) ? dst[k] : (k - E);
    float ex = expf(e[k] - fdec(mkey[d]));
    e[k] = ex;
    atomicAdd(&denom[d], ex);
}

__launch_bounds__(256)
__global__ void edge_agg(const float* __restrict__ e,
                         const float4* __restrict__ hl4,
                         const int* __restrict__ src,
                         const int* __restrict__ dst,
                         float* __restrict__ agg,
                         int E, int EN) {
    int w    = blockIdx.x * (blockDim.x >> 5) + (threadIdx.x >> 5);
    int lane = threadIdx.x & 31;
    if (w >= EN) return;
    int s = (w < E) ? src[w] : (w - E);
    int d = (w < E) ? dst[w] : (w - E);
    float ex  = e[w];
    float4 hv = hl4[(size_t)s * 32 + lane];
    float* base = agg + (size_t)d * HIDC + lane * 4;
    atomicAdd(base + 0, ex * hv.x);
    atomicAdd(base + 1, ex * hv.y);
    atomicAdd(base + 2, ex * hv.z);
    atomicAdd(base + 3, ex * hv.w);
}

// finalize: h = agg/denom + bias ; accumulate BN sum / sumsq (1 atomic/chan/blk)
__launch_bounds__(128)
__global__ void node_finalize(float* __restrict__ h,
                              const float* __restrict__ denom,
                              const float* __restrict__ bias,
                              float* __restrict__ bnsum,
                              float* __restrict__ bnsq,
                              int N, int doBN) {
    int c  = threadIdx.x;
    int n0 = blockIdx.x * 64;
    int rows = min(64, N - n0);
    float bc = bias[c];
    float s = 0.0f, s2 = 0.0f;
    for (int i = 0; i < rows; ++i) {
        int n = n0 + i;
        float v = h[(size_t)n * HIDC + c] / denom[n] + bc;
        h[(size_t)n * HIDC + c] = v;
        s += v;
        s2 += v * v;
    }
    if (doBN && rows > 0) {
        atomicAdd(&bnsum[c], s);
        atomicAdd(&bnsq[c], s2);
    }
}

__launch_bounds__(128)
__global__ void bn_apply(float* __restrict__ h,
                         const float* __restrict__ bnsum,
                         const float* __restrict__ bnsq,
                         const float* __restrict__ gamma,
                         const float* __restrict__ beta,
                         int N) {
    int c = threadIdx.x;
    float invN = 1.0f / (float)N;
    float mu  = bnsum[c] * invN;
    float var = bnsq[c] * invN - mu * mu;
    float g   = gamma[c] * rsqrtf(var + 1e-5f);
    float bo  = beta[c] - mu * g;
    int n0 = blockIdx.x * 64;
    int rows = min(64, N - n0);
    for (int i = 0; i < rows; ++i) {
        size_t idx = (size_t)(n0 + i) * HIDC + c;
        float v = h[idx] * g + bo;
        h[idx] = v > 0.0f ? v : 0.01f * v;
    }
}

// ---------------------------------------------------------------------------
// Graph pooling (mean + max per graph)
// ---------------------------------------------------------------------------
__launch_bounds__(128)
__global__ void pool_accum(const float* __restrict__ h,
                           const int* __restrict__ batch,
                           float* __restrict__ p1,
                           unsigned* __restrict__ pmax,
                           float* __restrict__ cnt,
                           int N) {
    int c  = threadIdx.x;
    int n0 = blockIdx.x * 64;
    int rows = min(64, N - n0);
    for (int i = 0; i < rows; ++i) {
        int n = n0 + i;
        int b = batch[n];
        float v = h[(size_t)n * HIDC + c];
        atomicAdd(&p1[(size_t)b * HIDC + c], v);
        atomicMax(&pmax[(size_t)b * HIDC + c], fenc(v));
        if (c == 0) atomicAdd(&cnt[b], 1.0f);
    }
}

__global__ void pool_final(const float* __restrict__ p1,
                           const unsigned* __restrict__ pmax,
                           const float* __restrict__ cnt,
                           float* __restrict__ z, int G) {
    int i = blockIdx.x * blockDim.x + threadIdx.x;
    if (i >= G * HIDC) return;
    int g = i >> 7, c = i & 127;
    float ct   = cnt[g];
    float mean = p1[i] / fmaxf(ct, 1.0f);
    float mx   = ct > 0.0f ? fdec(pmax[i]) : 0.0f;
    z[(size_t)g * 256 + c]       = mean;
    z[(size_t)g * 256 + 128 + c] = mx;
}

__global__ void fc2_kernel(const float* __restrict__ x1,
                           const float* __restrict__ W,   // [64,3]
                           const float* __restrict__ b,
                           float* __restrict__ out, int G) {
    int i = blockIdx.x * blockDim.x + threadIdx.x;
    if (i >= G * 3) return;
    int g = i / 3, j = i - g * 3;
    const float* xr = x1 + (size_t)g * 64;
    float s = b[j];
    for (int k = 0; k < 64; ++k) s += xr[k] * W[k * 3 + j];
    out[i] = s;
}

__global__ void fill_f32(float* p, float v, size_t n) {
    size_t i = (size_t)blockIdx.x * blockDim.x + threadIdx.x;
    if (i < n) p[i] = v;
}
__global__ void fill_u32(unsigned* p, unsigned v, size_t n) {
    size_t i = (size_t)blockIdx.x * blockDim.x + threadIdx.x;
    if (i < n) p[i] = v;
}

// ---------------------------------------------------------------------------
extern "C" void kernel_launch(void* const* d_in, const int* in_sizes, int n_in,
                              void* d_out, int out_size, void* d_ws, size_t ws_size,
                              hipStream_t stream) {
    (void)n_in; (void)ws_size;
    const int N  = in_sizes[0] / 48;
    const int E  = in_sizes[1];
    const int EN = E + N;
    const int G  = out_size / 67;  // out[G*3] ++ x1[G*64]

    const float* x     = (const float*)d_in[0];
    const int*   src   = (const int*)d_in[1];
    const int*   dst   = (const int*)d_in[2];
    const int*   batch = (const int*)d_in[3];
    const float* Wl0   = (const float*)d_in[4];
    const float* Wr0   = (const float*)d_in[5];
    const float* att0  = (const float*)d_in[6];
    const float* b0    = (const float*)d_in[7];
    const float* Wl    = (const float*)d_in[8];
    const float* Wr    = (const float*)d_in[9];
    const float* att   = (const float*)d_in[10];
    const float* bb    = (const float*)d_in[11];
    const float* gamma = (const float*)d_in[12];
    const float* beta  = (const float*)d_in[13];
    const float* fcW   = (const float*)d_in[14];
    const float* fcb   = (const float*)d_in[15];
    const float* fc2W  = (const float*)d_in[16];
    const float* fc2b  = (const float*)d_in[17];

    // scratch carve (chunks unpadded; hl/hr, read as float4, start at multiples of 128 floats -> 16B aligned)
    float* ws = (float*)d_ws;
    size_t off = 0;
    auto carve = [&](size_t n) { float* p = ws + off; off += n; return p; };
    float*    hA    = carve((size_t)N * HIDC);
    float*    hB    = carve((size_t)N * HIDC);
    float*    hl    = carve((size_t)N * HIDC);
    float*    hr    = carve((size_t)N * HIDC);
    float*    e     = carve((size_t)EN);
    float*    denom = carve((size_t)N);
    unsigned* mkey  = (unsigned*)carve((size_t)N);
    float*    bnsum = carve(128);
    float*    bnsq  = carve(128);
    float*    p1    = carve((size_t)G * HIDC);
    unsigned* pmax  = (unsigned*)carve((size_t)G * HIDC);
    float*    cnt   = carve((size_t)G);
    float*    z     = carve((size_t)G * 256);

    const int nodeBlocks = (N + 63) / 64;
    const int edgeWBlks  = (EN + 7) / 8;     // 8 waves / 256-thread block
    auto fb = [](size_t n) { return (int)((n + 255) / 256); };

    const float* cur = x;
    float* bufs[2] = {hA, hB};
    int pb = 0;

    for (int L = 0; L < 6; ++L) {
        const float* wl = (L == 0) ? Wl0  : Wl  + (size_t)(L - 1) * HIDC * HIDC;
        const float* wr = (L == 0) ? Wr0  : Wr  + (size_t)(L - 1) * HIDC * HIDC;
        const float* a  = (L == 0) ? att0 : att + (size_t)(L - 1) * HIDC;
        const float* bi = (L == 0) ? b0   : bb  + (size_t)(L - 1) * HIDC;
        float* agg = bufs[pb];

        if (L == 0) {
            gemm_f32_wmma<48, 128><<<nodeBlocks, 128, 0, stream>>>(cur, wl, nullptr, hl, N, 0);
            gemm_f32_wmma<48, 128><<<nodeBlocks, 128, 0, stream>>>(cur, wr, nullptr, hr, N, 0);
        } else {
            gemm_f32_wmma<128, 128><<<nodeBlocks, 128, 0, stream>>>(cur, wl, nullptr, hl, N, 0);
            gemm_f32_wmma<128, 128><<<nodeBlocks, 128, 0, stream>>>(cur, wr, nullptr, hr, N, 0);
        }

        fill_f32<<<fb((size_t)N * HIDC), 256, 0, stream>>>(agg, 0.0f, (size_t)N * HIDC);
        fill_f32<<<fb(N), 256, 0, stream>>>(denom, 0.0f, (size_t)N);
        fill_u32<<<fb(N), 256, 0, stream>>>(mkey, NEG_INF_KEY, (size_t)N);
        fill_f32<<<1, 256, 0, stream>>>(bnsum, 0.0f, 256);  // bnsum+bnsq contiguous

        edge_score<<<edgeWBlks, 256, 0, stream>>>((const float4*)hl, (const float4*)hr,
                                                  (const float4*)a, src, dst, e, mkey, E, EN);
        edge_exp<<<fb(EN), 256, 0, stream>>>(dst, e, mkey, denom, E, EN);
        edge_agg<<<edgeWBlks, 256, 0, stream>>>(e, (const float4*)hl, src, dst, agg, E, EN);

        int doBN = (L < 5);
        node_finalize<<<nodeBlocks, 128, 0, stream>>>(agg, denom, bi, bnsum, bnsq, N, doBN);
        if (doBN)
            bn_apply<<<nodeBlocks, 128, 0, stream>>>(agg, bnsum, bnsq,
                                                     gamma + (size_t)L * HIDC,
                                                     beta + (size_t)L * HIDC, N);
        cur = agg;
        pb ^= 1;
    }

    // pooling
    fill_f32<<<fb((size_t)G * HIDC), 256, 0, stream>>>(p1, 0.0f, (size_t)G * HIDC);
    fill_u32<<<fb((size_t)G * HIDC), 256, 0, stream>>>(pmax, NEG_INF_KEY, (size_t)G * HIDC);
    fill_f32<<<fb(G), 256, 0, stream>>>(cnt, 0.0f, (size_t)G);
    pool_accum<<<nodeBlocks, 128, 0, stream>>>(cur, batch, p1, pmax, cnt, N);
    pool_final<<<fb((size_t)G * HIDC), 256, 0, stream>>>(p1, pmax, cnt, z, G);

    // head: x1 = lrelu(z @ fcW + fcb) -> d_out[G*3 ..]; out = x1 @ fc2W + fc2b
    float* outp = (float*)d_out;
    float* x1   = outp + (size_t)G * 3;
    gemm_f32_wmma<256, 64><<<(G + 63) / 64, 128, 0, stream>>>(z, fcW, fcb, x1, G, 1);
    fc2_kernel<<<fb((size_t)G * 3), 256, 0, stream>>>(x1, fc2W, fc2b, outp, G);
}